// PreProcess_44616120271335
// MI455X (gfx1250) — compile-verified
//
#include <hip/hip_runtime.h>

typedef __attribute__((ext_vector_type(2))) float v2f;
typedef __attribute__((ext_vector_type(8))) float v8f;

#define B_ 32
#define P_ 128
#define D_ 512
#define EPS_ 1e-8f

// ws layout (floats):
//   [0, 4096)             rownorm2[row],  row = b*128 + p  (matches x row order)
//   [4096, 4224)          norms2[p]
//   [4224, 4224+131072)   G[L*32 + j] = x_row(L) . x[0, 4j, :]

// ---------------------------------------------------------------------------
// 1) per-row squared norms: one wave32 per row of x (4096 rows x 512 elems)
__global__ void rownorm_kernel(const float* __restrict__ x, float* __restrict__ rn2) {
    const int wave = (blockIdx.x * blockDim.x + threadIdx.x) >> 5;
    const int lane = threadIdx.x & 31;
    if (wave >= B_ * P_) return;
    const float* row = x + (size_t)wave * D_;
    float s = 0.f;
    #pragma unroll 4
    for (int i = lane; i < D_; i += 32) { const float v = row[i]; s += v * v; }
    #pragma unroll
    for (int off = 16; off > 0; off >>= 1) s += __shfl_down(s, off, 32);
    if (lane == 0) rn2[wave] = s;
}

// ---------------------------------------------------------------------------
// 2) norms2[p] = sum_b rownorm2[b*128 + p]
__global__ void colnorm_kernel(const float* __restrict__ rn2, float* __restrict__ n2) {
    const int p = threadIdx.x;
    if (p >= P_) return;
    float s = 0.f;
    #pragma unroll
    for (int b = 0; b < B_; ++b) s += rn2[b * P_ + p];
    n2[p] = s;
}

// ---------------------------------------------------------------------------
// 3) fp32 WMMA GEMM: G[L][j] = x_row(L) . x[0, 4j, :]
//    x_row(L) = x[L%32, L//32, :].  One wave per 16x16 output tile.
//    A 16x4 f32 layout: lanes 0-15 -> M=lane, K=k0+{0,1}; lanes 16-31 -> K=k0+{2,3}.
//    B 4x16 layout mirrors (VGPR0: K=k0+koff, VGPR1: K=k0+koff+1; N=lane&15).
__global__ void __launch_bounds__(32) gemm_wmma_kernel(const float* __restrict__ x,
                                                       float* __restrict__ G) {
    const int mt   = blockIdx.x;        // 0..255  (16-row tile of L)
    const int nt   = blockIdx.y;        // 0..1    (16-col tile of j)
    const int lane = threadIdx.x;       // 0..31
    const int r    = lane & 15;
    const int koff = (lane >> 4) * 2;

    // Within this 16-row tile: p is constant, bb spans a contiguous 16-range.
    const int p  = mt >> 1;
    const int bb = (mt & 1) * 16 + r;
    const int j  = nt * 16 + r;

    const float* arow = x + ((size_t)bb * P_ + p) * D_;  // A row for this lane's M
    const float* brow = x + (size_t)(4 * j) * D_;        // B col j = x[0, 4j, :]

    v8f acc = {};
    for (int k0 = 0; k0 < D_; k0 += 4) {
        v2f a, b;
        a[0] = arow[k0 + koff + 0];
        a[1] = arow[k0 + koff + 1];
        b[0] = brow[k0 + koff + 0];
        b[1] = brow[k0 + koff + 1];
        // D = A(16x4,f32) * B(4x16,f32) + C ; full fp32 precision
        acc = __builtin_amdgcn_wmma_f32_16x16x4_f32(false, a, false, b,
                                                    (short)0, acc, false, false);
    }

    // C/D layout: VGPR i -> M = i + (lane>=16 ? 8 : 0), N = lane&15
    const int rowBase = 16 * mt + (lane >> 4) * 8;
    const int col     = nt * 16 + r;
    #pragma unroll
    for (int i = 0; i < 8; ++i)
        G[(size_t)(rowBase + i) * 32 + col] = acc[i];
}

// ---------------------------------------------------------------------------
// 4) d[b,k] = cos(u[b,k], u[b,k]+u[b,0]) via normalized dots; out[b+32k] = mean_k d
__global__ void finalize_kernel(const float* __restrict__ rn2,
                                const float* __restrict__ n2,
                                const float* __restrict__ G,
                                float* __restrict__ out) {
    __shared__ float sm[P_];
    const int b = blockIdx.x;     // 0..31
    const int k = threadIdx.x;    // 0..127
    const int L = b * P_ + k;
    const int p  = L >> 5;        // L // 32
    const int bb = L & 31;        // L % 32

    const float n2p = n2[p];
    const float n2c = n2[4 * b];
    const float aa  = rn2[bb * P_ + p] / n2p;                         // |a|^2
    const float cc  = rn2[4 * b] / n2c;                               // |c|^2  (row bb=0, p=4b)
    const float ac  = G[(size_t)L * 32 + b] / (sqrtf(n2p) * sqrtf(n2c));

    const float num = 0.5f * (aa + ac);
    const float den = 0.5f * sqrtf(aa) * sqrtf(fmaxf(aa + 2.f * ac + cc, 0.f));
    float dv = num / fmaxf(den, EPS_);

    sm[k] = dv;
    __syncthreads();
    #pragma unroll
    for (int s = 64; s > 0; s >>= 1) {
        if (k < s) sm[k] += sm[k + s];
        __syncthreads();
    }
    const float sb = sm[0] * (1.f / (float)P_);
    out[b + B_ * k] = sb;   // covers every L' with L'%32 == b
}

// ---------------------------------------------------------------------------
extern "C" void kernel_launch(void* const* d_in, const int* in_sizes, int n_in,
                              void* d_out, int out_size, void* d_ws, size_t ws_size,
                              hipStream_t stream) {
    (void)in_sizes; (void)n_in; (void)out_size; (void)ws_size;
    const float* x  = (const float*)d_in[0];
    float*       ws = (float*)d_ws;
    float* rn2 = ws;                    // 4096 floats
    float* n2  = ws + 4096;             // 128 floats
    float* G   = ws + 4096 + 128;       // 131072 floats
    float* out = (float*)d_out;         // 4096 floats

    rownorm_kernel<<<dim3(512), dim3(256), 0, stream>>>(x, rn2);
    colnorm_kernel<<<dim3(1), dim3(128), 0, stream>>>(rn2, n2);
    gemm_wmma_kernel<<<dim3(256, 2), dim3(32), 0, stream>>>(x, G);
    finalize_kernel<<<dim3(32), dim3(128), 0, stream>>>(rn2, n2, G, out);
}